// EnhancedSetOfSetBlock_44659069943822
// MI455X (gfx1250) — compile-verified
//
#include <hip/hip_runtime.h>
#include <hip/hip_bf16.h>
#include <math.h>

// ---------------------------------------------------------------------------
// EnhancedSetOfSetBlock forward for MI455X (gfx1250, wave32, WMMA).
// Dominant work = fp32 GEMMs done with V_WMMA_F32_16X16X4_F32 (full fp32
// precision, matching the fp32 reference). LDS-tiled, 4 waves/block, each
// wave owns a 16x64 output strip (4 x v8f accumulators). Both WMMA operands
// are stored in LDS so each {k, k+1} pair is a single aligned ds_load_b64:
//   sA : row-major, row stride BK+2 (8B-aligned pairs along K)
//   sWt: TRANSPOSED (col-major panel), same stride, so B pairs are contiguous
// ---------------------------------------------------------------------------

typedef float v2f __attribute__((ext_vector_type(2)));
typedef float v8f __attribute__((ext_vector_type(8)));

#define GEMM_BM 64
#define GEMM_BK 32
#define GEMM_LDA (GEMM_BK + 2)   // pad 2: keeps 8-byte alignment of (row, even k)
#define GEMM_THREADS 128         // 4 waves of 32

// C[M,N] = A[M,K] @ W[K,N] + bias[N]   (row major; N%64==0, K%32==0, M any)
__global__ __launch_bounds__(GEMM_THREADS)
void gemm_bias_wmma(const float* __restrict__ A, const float* __restrict__ W,
                    const float* __restrict__ bias, float* __restrict__ C,
                    int M, int N, int K) {
    __shared__ float sA [GEMM_BM][GEMM_LDA];   // A panel, K contiguous
    __shared__ float sWt[64]     [GEMM_LDA];   // W panel transposed: [col][k]

    const int tid  = threadIdx.x;
    const int wave = tid >> 5;
    const int lane = tid & 31;
    const int half = lane >> 4;     // 0: lanes 0-15, 1: lanes 16-31
    const int lo   = lane & 15;
    const int rowBase = blockIdx.x * GEMM_BM;
    const int waveRow = wave * 16;

    // Loop N-groups inside the block so this block's A rows stay L2-hot.
    for (int ng = 0; ng < N; ng += 64) {
        v8f acc[4];
        const v8f vzero = {0.f,0.f,0.f,0.f,0.f,0.f,0.f,0.f};
#pragma unroll
        for (int t = 0; t < 4; ++t) acc[t] = vzero;

        for (int k0 = 0; k0 < K; k0 += GEMM_BK) {
            // Stage A panel 64x32 (global reads coalesced along K)
            for (int i = tid; i < GEMM_BM * GEMM_BK; i += GEMM_THREADS) {
                int r  = i >> 5;          // /GEMM_BK
                int kk = i & (GEMM_BK - 1);
                int gr = rowBase + r;
                sA[r][kk] = (gr < M) ? A[(long long)gr * K + k0 + kk] : 0.f;
            }
            // Stage W panel 32x64 transposed (global reads coalesced along N)
            for (int i = tid; i < GEMM_BK * 64; i += GEMM_THREADS) {
                int kk = i >> 6;          // /64
                int c  = i & 63;
                sWt[c][kk] = W[(long long)(k0 + kk) * N + ng + c];
            }
            __syncthreads();

            // 16x16x4 fp32 WMMA micro-kernel.
            // A VGPRs: lanes 0-15 -> rows, K = kk+{0,1}; lanes 16-31 -> K = kk+{2,3}
            // B VGPRs: lanes 0-15 -> cols, K = kk+{0,1}; lanes 16-31 -> K = kk+{2,3}
#pragma unroll
            for (int kk = 0; kk < GEMM_BK; kk += 4) {
                const int kp = kk + 2 * half;          // even -> 8B aligned
                v2f a = *(const v2f*)(&sA[waveRow + lo][kp]);
#pragma unroll
                for (int t = 0; t < 4; ++t) {
                    v2f b = *(const v2f*)(&sWt[t * 16 + lo][kp]);
                    acc[t] = __builtin_amdgcn_wmma_f32_16x16x4_f32(
                        false, a, false, b, (short)0, acc[t], false, false);
                }
            }
            __syncthreads();
        }

        // Epilogue: C layout — v8f element r: row = r + 8*half, col = lane%16.
#pragma unroll
        for (int t = 0; t < 4; ++t) {
            int col = ng + t * 16 + lo;
            float bv = (bias != nullptr) ? bias[col] : 0.f;
#pragma unroll
            for (int r = 0; r < 8; ++r) {
                int grow = rowBase + waveRow + r + 8 * half;
                if (grow < M) C[(long long)grow * N + col] = acc[t][r] + bv;
            }
        }
    }
}

// ---------------------------------------------------------------------------
// Elementwise / reduction kernels
// ---------------------------------------------------------------------------

__global__ void fill_zero_kernel(float* p, long long n) {
    long long stride = (long long)gridDim.x * blockDim.x;
    for (long long i = (long long)blockIdx.x * blockDim.x + threadIdx.x; i < n; i += stride)
        p[i] = 0.f;
}

// Segment sums over edges into per-point and per-camera accumulators.
__global__ void seg_sums_kernel(const float* __restrict__ v,
                                const int* __restrict__ cam_idx,
                                const int* __restrict__ pt_idx,
                                float* __restrict__ cam_sum,
                                float* __restrict__ pt_sum,
                                long long total, int d) {
    long long stride = (long long)gridDim.x * blockDim.x;
    for (long long i = (long long)blockIdx.x * blockDim.x + threadIdx.x; i < total; i += stride) {
        long long e = i / d;
        int c = (int)(i - e * d);
        float x = v[i];
        atomicAdd(&pt_sum[(long long)pt_idx[e] * d + c], x);
        atomicAdd(&cam_sum[(long long)cam_idx[e] * d + c], x);
    }
}

__global__ void scale_rows_inv_count_kernel(float* m, const int* cnt, int rows, int d) {
    long long total = (long long)rows * d;
    long long stride = (long long)gridDim.x * blockDim.x;
    for (long long i = (long long)blockIdx.x * blockDim.x + threadIdx.x; i < total; i += stride) {
        int r = (int)(i / d);
        int c = cnt[r];
        m[i] = (c > 0) ? m[i] / (float)c : 0.f;
    }
}

__global__ void scale_const_kernel(float* m, float s, long long n) {
    long long stride = (long long)gridDim.x * blockDim.x;
    for (long long i = (long long)blockIdx.x * blockDim.x + threadIdx.x; i < n; i += stride)
        m[i] *= s;
}

// out[e,c] = out_all[e,c] + out_rows[pt[e],c] + out_cols[cam[e],c] + out_both[c]
__global__ void combine_sos_kernel(const float* __restrict__ out_all,
                                   const float* __restrict__ out_rows,
                                   const float* __restrict__ out_cols,
                                   const float* __restrict__ out_both,
                                   const int* __restrict__ cam_idx,
                                   const int* __restrict__ pt_idx,
                                   float* __restrict__ out,
                                   long long total, int d) {
    long long stride = (long long)gridDim.x * blockDim.x;
    for (long long i = (long long)blockIdx.x * blockDim.x + threadIdx.x; i < total; i += stride) {
        long long e = i / d;
        int c = (int)(i - e * d);
        out[i] = out_all[i] + out_rows[(long long)pt_idx[e] * d + c]
                            + out_cols[(long long)cam_idx[e] * d + c] + out_both[c];
    }
}

// g[c] = mean over rows of m[r,c]   (one block, d threads)
__global__ void col_mean_small_kernel(const float* __restrict__ m, float* __restrict__ out,
                                      int rows, int d) {
    int c = threadIdx.x;
    if (c >= d) return;
    float s = 0.f;
    for (int r = 0; r < rows; ++r) s += m[(long long)r * d + c];
    out[c] = s / (float)rows;
}

// Column stats over [nnz, d]: deterministic two-pass (no atomics).
__global__ void colstats_partial_kernel(const float* __restrict__ v, float* ps, float* pss,
                                        int nnz, int d, int rowsPerBlk) {
    int c = threadIdx.x;           // blockDim.x == d
    int r0 = blockIdx.x * rowsPerBlk;
    int r1 = r0 + rowsPerBlk; if (r1 > nnz) r1 = nnz;
    float s = 0.f, ss = 0.f;
    for (int r = r0; r < r1; ++r) {
        float x = v[(long long)r * d + c];
        s += x; ss += x * x;
    }
    ps[(long long)blockIdx.x * d + c] = s;
    pss[(long long)blockIdx.x * d + c] = ss;
}

__global__ void colstats_final_kernel(const float* __restrict__ ps, const float* __restrict__ pss,
                                      float* __restrict__ stats, int nblk, int d, int nnz) {
    int c = threadIdx.x;
    if (c >= d) return;
    float s = 0.f, ss = 0.f;
    for (int b = 0; b < nblk; ++b) { s += ps[(long long)b * d + c]; ss += pss[(long long)b * d + c]; }
    float mean = s / (float)nnz;
    float var1 = (ss - (float)nnz * mean * mean) / (float)(nnz - 1);   // ddof=1
    var1 = fmaxf(var1, 1e-20f);
    stats[c] = mean;
    stats[d + c] = rsqrtf(var1);
}

__global__ void apply_colnorm_kernel(float* v, const float* __restrict__ stats,
                                     long long total, int d, int doRelu) {
    long long stride = (long long)gridDim.x * blockDim.x;
    for (long long i = (long long)blockIdx.x * blockDim.x + threadIdx.x; i < total; i += stride) {
        int c = (int)(i % d);
        float x = (v[i] - stats[c]) * stats[d + c];
        if (doRelu) x = fmaxf(x, 0.f);
        v[i] = x;
    }
}

// ---------------------------------------------------------------------------
// Attention: one block per (query row, head). Scores live in LDS.
// ---------------------------------------------------------------------------
__global__ void attn_core_kernel(const float* __restrict__ Q, const float* __restrict__ Kk,
                                 const float* __restrict__ Vv, float* __restrict__ ctx,
                                 int Sk, int d, int dk, float scale) {
    extern __shared__ float smem[];
    float* p   = smem;            // [Sk]  scores / probs
    float* qv  = smem + Sk;       // [dk]
    float* red = qv + dk;         // [blockDim.x]
    const int q = blockIdx.x, h = blockIdx.y;
    const int tid = threadIdx.x, nt = blockDim.x;
    const int hoff = h * dk;

    if (tid < dk) qv[tid] = Q[(long long)q * d + hoff + tid];
    __syncthreads();

    // scores + max
    float lmax = -3.0e38f;
    for (int k = tid; k < Sk; k += nt) {
        const float* kr = Kk + (long long)k * d + hoff;
        float s = 0.f;
        for (int j = 0; j < dk; ++j) s += qv[j] * kr[j];
        s *= scale;
        p[k] = s;
        lmax = fmaxf(lmax, s);
    }
    red[tid] = lmax; __syncthreads();
    for (int off = nt >> 1; off > 0; off >>= 1) {
        if (tid < off) red[tid] = fmaxf(red[tid], red[tid + off]);
        __syncthreads();
    }
    float mx = red[0]; __syncthreads();

    // exp + sum
    float lsum = 0.f;
    for (int k = tid; k < Sk; k += nt) { float e = __expf(p[k] - mx); p[k] = e; lsum += e; }
    red[tid] = lsum; __syncthreads();
    for (int off = nt >> 1; off > 0; off >>= 1) {
        if (tid < off) red[tid] += red[tid + off];
        __syncthreads();
    }
    float inv = 1.f / red[0]; __syncthreads();

    // ctx[dd] = sum_k p[k] * V[k, hoff+dd]; threads = (group g) x (dim dd)
    int dd = tid % dk;
    int g  = tid / dk;
    int ng = nt / dk;
    float accv = 0.f;
    for (int k = g; k < Sk; k += ng) accv += p[k] * Vv[(long long)k * d + hoff + dd];
    red[g * dk + dd] = accv; __syncthreads();
    if (g == 0) {
        float s = 0.f;
        for (int gg = 0; gg < ng; ++gg) s += red[gg * dk + dd];
        ctx[(long long)q * d + hoff + dd] = s * inv;
    }
}

// out[r,:] = LayerNorm(x[r,:] + resid[r,:]) * g + b   (block per row, blockDim == d)
__global__ void ln_residual_kernel(const float* __restrict__ x, const float* __restrict__ resid,
                                   const float* __restrict__ g, const float* __restrict__ b,
                                   float* __restrict__ out, int d, float eps) {
    __shared__ float red[256];
    int row = blockIdx.x, c = threadIdx.x;
    float v = x[(long long)row * d + c] + resid[(long long)row * d + c];
    red[c] = v; __syncthreads();
    for (int off = d >> 1; off > 0; off >>= 1) {
        if (c < off) red[c] += red[c + off];
        __syncthreads();
    }
    float mean = red[0] / (float)d; __syncthreads();
    float dv = v - mean;
    red[c] = dv * dv; __syncthreads();
    for (int off = d >> 1; off > 0; off >>= 1) {
        if (c < off) red[c] += red[c + off];
        __syncthreads();
    }
    float var = red[0] / (float)d;   // ddof=0 + eps, as in reference
    out[(long long)row * d + c] = dv * rsqrtf(var + eps) * g[c] + b[c];
}

// final: relu(values + v2 + 0.5*(enh_cam[cam] + enh_pt[pt]))
__global__ void final_out_kernel(const float* __restrict__ values, const float* __restrict__ v2,
                                 const float* __restrict__ enh_cam, const float* __restrict__ enh_pt,
                                 const int* __restrict__ cam_idx, const int* __restrict__ pt_idx,
                                 float* __restrict__ out, long long total, int d) {
    long long stride = (long long)gridDim.x * blockDim.x;
    for (long long i = (long long)blockIdx.x * blockDim.x + threadIdx.x; i < total; i += stride) {
        long long e = i / d;
        int c = (int)(i - e * d);
        float x = values[i] + v2[i]
                + 0.5f * (enh_cam[(long long)cam_idx[e] * d + c]
                        + enh_pt[(long long)pt_idx[e] * d + c]);
        out[i] = fmaxf(x, 0.f);
    }
}

// ---------------------------------------------------------------------------
// Host orchestration
// ---------------------------------------------------------------------------

static inline int gs_blocks(long long n) {
    long long b = (n + 255) / 256;
    if (b > 16384) b = 16384;
    if (b < 1) b = 1;
    return (int)b;
}

extern "C" void kernel_launch(void* const* d_in, const int* in_sizes, int n_in,
                              void* d_out, int out_size, void* d_ws, size_t ws_size,
                              hipStream_t stream) {
    const float* values   = (const float*)d_in[0];
    const float* sos_W    = (const float*)d_in[1];   // [2,4,D,D]
    const float* sos_b    = (const float*)d_in[2];   // [2,4,D]
    const float* self_W   = (const float*)d_in[3];   // [4,D,D]
    const float* self_b   = (const float*)d_in[4];   // [4,D]
    const float* self_ln  = (const float*)d_in[5];   // [2,D]
    const float* cross_W  = (const float*)d_in[6];
    const float* cross_b  = (const float*)d_in[7];
    const float* cross_ln = (const float*)d_in[8];
    const int* cam_idx    = (const int*)d_in[9];
    const int* pt_idx     = (const int*)d_in[10];
    const int* cam_per_pt = (const int*)d_in[11];
    const int* pts_per_cam= (const int*)d_in[12];

    const int NNZ = in_sizes[9];
    const int NP  = in_sizes[11];          // n_pts  (length of cam_per_pt)
    const int NC  = in_sizes[12];          // n_cams (length of pts_per_cam)
    const int D   = in_sizes[0] / NNZ;     // 256
    const int H   = 8;
    const int DK  = D / H;
    const long long EDG = (long long)NNZ * D;
    const float attn_scale = 1.f / sqrtf((float)DK);

    // ---- carve workspace (floats) ----
    float* ws = (float*)d_ws;
    long long off = 0;
    auto take = [&](long long n) { float* p = ws + off; off += n; return p; };
    const int statRows = 512;
    const int nblkStats = (NNZ + statRows - 1) / statRows;

    float* bufA    = take(EDG);                  // out_all / out_all2
    float* bufB    = take(EDG);                  // v1 then v2
    float* ptS     = take((long long)NP * D);    // point sums -> means / pt_feat
    float* camS    = take((long long)NC * D);    // cam sums -> means / cam_feat
    float* outR    = take((long long)NP * D);
    float* outC    = take((long long)NC * D);
    float* gmean   = take(D);
    float* outBoth = take(D);
    float* camF2   = take((long long)NC * D);    // self-attended cam feat
    float* Qb      = take((long long)NP * D);
    float* Kb      = take((long long)NP * D);
    float* Vb      = take((long long)NP * D);
    float* ctxB    = take((long long)NP * D);
    float* projB   = take((long long)NP * D);
    float* enhC    = take((long long)NC * D);
    float* enhP    = take((long long)NP * D);
    float* ps      = take((long long)nblkStats * D);
    float* pss     = take((long long)nblkStats * D);
    float* stats   = take(2 * D);
    (void)ws_size; (void)n_in; (void)out_size;

    auto gemm = [&](const float* A, const float* W, const float* b, float* C,
                    int M, int N, int K) {
        dim3 grid((M + GEMM_BM - 1) / GEMM_BM);
        gemm_bias_wmma<<<grid, GEMM_THREADS, 0, stream>>>(A, W, b, C, M, N, K);
    };

    // ---- one SetOfSet layer: X -> vout (uses bufA as out_all scratch) ----
    auto run_sos = [&](const float* X, const float* Wl, const float* bl, float* vout) {
        fill_zero_kernel<<<gs_blocks((long long)NP * D), 256, 0, stream>>>(ptS, (long long)NP * D);
        fill_zero_kernel<<<gs_blocks((long long)NC * D), 256, 0, stream>>>(camS, (long long)NC * D);
        seg_sums_kernel<<<gs_blocks(EDG), 256, 0, stream>>>(X, cam_idx, pt_idx, camS, ptS, EDG, D);
        scale_rows_inv_count_kernel<<<gs_blocks((long long)NP * D), 256, 0, stream>>>(ptS, cam_per_pt, NP, D);
        scale_rows_inv_count_kernel<<<gs_blocks((long long)NC * D), 256, 0, stream>>>(camS, pts_per_cam, NC, D);
        gemm(ptS,  Wl + 1LL * D * D, bl + 1LL * D, outR, NP, D, D);      // per-point
        gemm(camS, Wl + 2LL * D * D, bl + 2LL * D, outC, NC, D, D);      // per-camera
        col_mean_small_kernel<<<1, D, 0, stream>>>(camS, gmean, NC, D);
        gemm(gmean, Wl + 3LL * D * D, bl + 3LL * D, outBoth, 1, D, D);   // global
        gemm(X, Wl, bl, bufA, NNZ, D, D);                                 // per-edge (big)
        combine_sos_kernel<<<gs_blocks(EDG), 256, 0, stream>>>(
            bufA, outR, outC, outBoth, cam_idx, pt_idx, vout, EDG, D);
    };

    auto run_colnorm = [&](float* v, int doRelu) {
        colstats_partial_kernel<<<nblkStats, D, 0, stream>>>(v, ps, pss, NNZ, D, statRows);
        colstats_final_kernel<<<1, D, 0, stream>>>(ps, pss, stats, nblkStats, D, NNZ);
        apply_colnorm_kernel<<<gs_blocks(EDG), 256, 0, stream>>>(v, stats, EDG, D, doRelu);
    };

    // attention: out = LN(proj(attn(xq,xkv,xkv)) + xq)
    auto run_attn = [&](const float* xq, int Sq, const float* xkv, int Sk,
                        const float* Wt, const float* bt, const float* lnt, float* outb) {
        gemm(xq,  Wt + 0LL * D * D, bt + 0LL * D, Qb, Sq, D, D);
        gemm(xkv, Wt + 1LL * D * D, bt + 1LL * D, Kb, Sk, D, D);
        gemm(xkv, Wt + 2LL * D * D, bt + 2LL * D, Vb, Sk, D, D);
        dim3 ag(Sq, H);
        size_t shmem = (size_t)(Sk + DK + 256) * sizeof(float);
        attn_core_kernel<<<ag, 256, shmem, stream>>>(Qb, Kb, Vb, ctxB, Sk, D, DK, attn_scale);
        gemm(ctxB, Wt + 3LL * D * D, bt + 3LL * D, projB, Sq, D, D);
        ln_residual_kernel<<<Sq, D, 0, stream>>>(projB, xq, lnt, lnt + D, outb, D, 1e-5f);
    };

    // ================= forward pass =================
    // layer 1: SetOfSet -> Norm -> ReLU
    run_sos(values, sos_W, sos_b, bufB);
    run_colnorm(bufB, /*relu=*/1);
    // layer 2: SetOfSet -> Norm
    run_sos(bufB, sos_W + 4LL * D * D, sos_b + 4LL * D, bufB);
    run_colnorm(bufB, /*relu=*/0);

    // dense features: cam_feat = segsum/NP, pt_feat = segsum/NC
    fill_zero_kernel<<<gs_blocks((long long)NP * D), 256, 0, stream>>>(ptS, (long long)NP * D);
    fill_zero_kernel<<<gs_blocks((long long)NC * D), 256, 0, stream>>>(camS, (long long)NC * D);
    seg_sums_kernel<<<gs_blocks(EDG), 256, 0, stream>>>(bufB, cam_idx, pt_idx, camS, ptS, EDG, D);
    scale_const_kernel<<<gs_blocks((long long)NC * D), 256, 0, stream>>>(camS, 1.f / (float)NP, (long long)NC * D);
    scale_const_kernel<<<gs_blocks((long long)NP * D), 256, 0, stream>>>(ptS, 1.f / (float)NC, (long long)NP * D);

    // self-attention on cameras, then cross attention both ways
    run_attn(camS, NC, camS, NC, self_W, self_b, self_ln, camF2);     // cam_feat'
    run_attn(camF2, NC, ptS, NP, cross_W, cross_b, cross_ln, enhC);   // enh_cam
    run_attn(ptS, NP, camF2, NC, cross_W, cross_b, cross_ln, enhP);   // enh_pt

    // final gather-add + skip + relu
    final_out_kernel<<<gs_blocks(EDG), 256, 0, stream>>>(
        values, bufB, enhC, enhP, cam_idx, pt_idx, (float*)d_out, EDG, D);
}